// ADGCN_7232724927262
// MI455X (gfx1250) — compile-verified
//
#include <hip/hip_runtime.h>
#include <hip/hip_bf16.h>

typedef float v2f __attribute__((ext_vector_type(2)));
typedef float v8f __attribute__((ext_vector_type(8)));

// ---------------------------------------------------------------------------
// Degree / normalization helpers
// ---------------------------------------------------------------------------
__global__ void k_fill(float* __restrict__ p, float v, int n) {
    int i = blockIdx.x * blockDim.x + threadIdx.x;
    if (i < n) p[i] = v;
}

__global__ void k_deg_scatter(float* __restrict__ deg, const int* __restrict__ col, int E) {
    int e = blockIdx.x * blockDim.x + threadIdx.x;
    if (e < E)
        __hip_atomic_fetch_add(&deg[col[e]], 1.0f, __ATOMIC_RELAXED, __HIP_MEMORY_SCOPE_AGENT);
}

__global__ void k_rsqrt(float* __restrict__ p, int n) {
    int i = blockIdx.x * blockDim.x + threadIdx.x;
    if (i < n) p[i] = rsqrtf(p[i]);   // deg >= 1 always (self loop), no zero guard needed
}

// ---------------------------------------------------------------------------
// FP32 WMMA GEMM:  C[M,N] = A[M,K] @ B[K,N]
// Block tile 128x64, K staged in LDS in chunks of 32, 8 waves per block.
// Wave w computes rows [16w,16w+16) x all 4 column tiles (4x v8f accumulators).
// Uses V_WMMA_F32_16X16X4_F32 (exact fp32 — workload is memory bound, no need
// to trade precision for matrix throughput).
// Requires: K % 32 == 0, N % 64 == 0 (true for 512/256/64 here). M guarded.
// ---------------------------------------------------------------------------
#define BM 128
#define BN 64
#define BK 32

__global__ __launch_bounds__(256) void k_gemm_wmma_f32(
    const float* __restrict__ A, const float* __restrict__ B, float* __restrict__ C,
    int M, int K, int N)
{
    __shared__ float As[BM][BK + 1];   // +1 pad: avoid bank conflicts on row-major frag reads
    __shared__ float Bs[BK][BN + 1];

    const int tid  = threadIdx.x;
    const int wave = tid >> 5;
    const int lane = tid & 31;
    const int half = lane >> 4;   // 0: lanes 0-15, 1: lanes 16-31
    const int idx  = lane & 15;
    const int mBase = blockIdx.x * BM;
    const int nBase = blockIdx.y * BN;

    v8f acc[4] = {};  // zero-init accumulators (16x16 f32 C/D tiles)

    for (int k0 = 0; k0 < K; k0 += BK) {
        // Stage A chunk (128x32), coalesced, zero-padded beyond M
        #pragma unroll
        for (int i = tid; i < BM * BK; i += 256) {
            int r = i / BK, c = i % BK;
            int gr = mBase + r;
            As[r][c] = (gr < M) ? A[(size_t)gr * K + (k0 + c)] : 0.0f;
        }
        // Stage B chunk (32x64)
        #pragma unroll
        for (int i = tid; i < BK * BN; i += 256) {
            int r = i / BN, c = i % BN;
            Bs[r][c] = B[(size_t)(k0 + r) * N + (nBase + c)];
        }
        // Hint the next A chunk toward the caches (global_prefetch_b8)
        if (k0 + BK < K && tid < BM) {
            int gr = mBase + tid;
            if (gr < M) __builtin_prefetch(&A[(size_t)gr * K + (k0 + BK)], 0, 1);
        }
        __syncthreads();

        #pragma unroll
        for (int kk = 0; kk < BK; kk += 4) {
            // A fragment (16x4 f32): lanes 0-15 hold K=kk,kk+1; lanes 16-31 hold K=kk+2,kk+3
            const int kb = kk + half * 2;
            const int ar = wave * 16 + idx;
            v2f a;
            a.x = As[ar][kb];
            a.y = As[ar][kb + 1];
            #pragma unroll
            for (int nt = 0; nt < 4; ++nt) {
                // B fragment (4x16 f32), mirrored layout
                v2f b;
                b.x = Bs[kb][nt * 16 + idx];
                b.y = Bs[kb + 1][nt * 16 + idx];
                acc[nt] = __builtin_amdgcn_wmma_f32_16x16x4_f32(
                    false, a, false, b, (short)0, acc[nt], false, false);
            }
        }
        __syncthreads();
    }

    // Store C/D tiles: VGPR r -> row (r + 8*half), lane idx -> column
    #pragma unroll
    for (int nt = 0; nt < 4; ++nt) {
        #pragma unroll
        for (int r = 0; r < 8; ++r) {
            int gr = mBase + wave * 16 + r + half * 8;
            int gc = nBase + nt * 16 + idx;
            if (gr < M) C[(size_t)gr * N + gc] = acc[nt][r];
        }
    }
}

// ---------------------------------------------------------------------------
// acc[i,:] = dinv[i]^2 * h[i,:]     (self-loop term; also initializes acc)
// ---------------------------------------------------------------------------
__global__ void k_selfloop_init(float* __restrict__ acc, const float* __restrict__ h,
                                const float* __restrict__ dinv, int n, int F)
{
    size_t i = (size_t)blockIdx.x * blockDim.x + threadIdx.x;   // float4 index
    size_t total = ((size_t)n * F) >> 2;
    if (i >= total) return;
    int node = (int)((i * 4) / F);
    float s = dinv[node]; s *= s;
    float4 v = reinterpret_cast<const float4*>(h)[i];
    float4 o; o.x = s * v.x; o.y = s * v.y; o.z = s * v.z; o.w = s * v.w;
    reinterpret_cast<float4*>(acc)[i] = o;
}

// ---------------------------------------------------------------------------
// Edge scatter: acc[col,:] += dinv[row]*dinv[col] * h[row,:]
// One edge per wave; lanes cover features with float4 gathers + f32 atomics.
// ---------------------------------------------------------------------------
__global__ __launch_bounds__(256) void k_edge_scatter(
    float* __restrict__ acc, const float* __restrict__ h,
    const int* __restrict__ row, const int* __restrict__ col,
    const float* __restrict__ dinv, int E, int F)
{
    const int wave = threadIdx.x >> 5;
    const int lane = threadIdx.x & 31;
    const int e = blockIdx.x * 8 + wave;
    if (e >= E) return;
    const int r = row[e];
    const int c = col[e];
    const float nrm = dinv[r] * dinv[c];
    const float* __restrict__ src = h + (size_t)r * F;
    float* __restrict__ dst = acc + (size_t)c * F;
    for (int f = lane * 4; f < F; f += 128) {
        float4 v = *reinterpret_cast<const float4*>(src + f);
        __hip_atomic_fetch_add(dst + f + 0, nrm * v.x, __ATOMIC_RELAXED, __HIP_MEMORY_SCOPE_AGENT);
        __hip_atomic_fetch_add(dst + f + 1, nrm * v.y, __ATOMIC_RELAXED, __HIP_MEMORY_SCOPE_AGENT);
        __hip_atomic_fetch_add(dst + f + 2, nrm * v.z, __ATOMIC_RELAXED, __HIP_MEMORY_SCOPE_AGENT);
        __hip_atomic_fetch_add(dst + f + 3, nrm * v.w, __ATOMIC_RELAXED, __HIP_MEMORY_SCOPE_AGENT);
    }
}

// ---------------------------------------------------------------------------
// In-place bias add (+ optional ReLU)
// ---------------------------------------------------------------------------
__global__ void k_bias_act(float* __restrict__ acc, const float* __restrict__ bias,
                           int n, int F, int relu)
{
    size_t i = (size_t)blockIdx.x * blockDim.x + threadIdx.x;   // float4 index
    size_t total = ((size_t)n * F) >> 2;
    if (i >= total) return;
    int f = (int)((i * 4) % F);
    float4 v = reinterpret_cast<float4*>(acc)[i];
    v.x += bias[f + 0]; v.y += bias[f + 1]; v.z += bias[f + 2]; v.w += bias[f + 3];
    if (relu) {
        v.x = fmaxf(v.x, 0.0f); v.y = fmaxf(v.y, 0.0f);
        v.z = fmaxf(v.z, 0.0f); v.w = fmaxf(v.w, 0.0f);
    }
    reinterpret_cast<float4*>(acc)[i] = v;
}

// ---------------------------------------------------------------------------
// In-place softmax over 64 classes; one wave32 per node (lane covers 2 elems)
// ---------------------------------------------------------------------------
__global__ __launch_bounds__(256) void k_softmax64(float* __restrict__ x, int n)
{
    const int wave = threadIdx.x >> 5;
    const int lane = threadIdx.x & 31;
    const int node = blockIdx.x * 8 + wave;
    if (node >= n) return;
    float* p = x + (size_t)node * 64;
    float a = p[lane];
    float b = p[lane + 32];
    float m = fmaxf(a, b);
    #pragma unroll
    for (int off = 16; off > 0; off >>= 1) m = fmaxf(m, __shfl_xor(m, off, 32));
    float ea = expf(a - m);
    float eb = expf(b - m);
    float s = ea + eb;
    #pragma unroll
    for (int off = 16; off > 0; off >>= 1) s += __shfl_xor(s, off, 32);
    float inv = 1.0f / s;
    p[lane]      = ea * inv;
    p[lane + 32] = eb * inv;
}

// ---------------------------------------------------------------------------
// Launch
// ---------------------------------------------------------------------------
extern "C" void kernel_launch(void* const* d_in, const int* in_sizes, int n_in,
                              void* d_out, int out_size, void* d_ws, size_t ws_size,
                              hipStream_t stream)
{
    const float* x  = (const float*)d_in[0];
    const int*   ei = (const int*)  d_in[1];
    const float* W1 = (const float*)d_in[2];
    const float* b1 = (const float*)d_in[3];
    const float* W2 = (const float*)d_in[4];
    const float* b2 = (const float*)d_in[5];
    const float* W3 = (const float*)d_in[6];
    const float* b3 = (const float*)d_in[7];

    const int HID = in_sizes[3];              // 256
    const int C   = in_sizes[5];              // 64
    const int FIN = in_sizes[2] / HID;        // 512
    const int N   = in_sizes[0] / FIN;        // 100000
    const int E   = in_sizes[1] / 2;          // 1.6M

    const int* rowp = ei;
    const int* colp = ei + E;

    // Workspace layout (reused across layers): two N x HID buffers + dinv
    char* ws = (char*)d_ws;
    size_t bufBytes = (size_t)N * HID * sizeof(float);
    float* bufA = (float*)ws;                    // h (GEMM output) of current layer
    float* bufB = (float*)(ws + bufBytes);       // aggregation accumulator
    float* dinv = (float*)(ws + 2 * bufBytes);   // N floats

    float* outp = (float*)d_out;                 // N x C

    const int T = 256;
    auto cdiv = [](long long a, long long b) { return (unsigned)((a + b - 1) / b); };

    // --- degrees (with self loops) and D^{-1/2} ---
    k_fill<<<cdiv(N, T), T, 0, stream>>>(dinv, 1.0f, N);
    k_deg_scatter<<<cdiv(E, T), T, 0, stream>>>(dinv, colp, E);
    k_rsqrt<<<cdiv(N, T), T, 0, stream>>>(dinv, N);

    // --- Layer 1: h1 = x @ W1 ; agg ; +b1 ; relu ---
    {
        dim3 grid(cdiv(N, BM), HID / BN);
        k_gemm_wmma_f32<<<grid, T, 0, stream>>>(x, W1, bufA, N, FIN, HID);
        k_selfloop_init<<<cdiv((long long)N * HID / 4, T), T, 0, stream>>>(bufB, bufA, dinv, N, HID);
        k_edge_scatter<<<cdiv(E, 8), T, 0, stream>>>(bufB, bufA, rowp, colp, dinv, E, HID);
        k_bias_act<<<cdiv((long long)N * HID / 4, T), T, 0, stream>>>(bufB, b1, N, HID, 1);
    }

    // --- Layer 2: h2 = agg1 @ W2 ; agg ; +b2 ; softmax ---
    {
        dim3 grid(cdiv(N, BM), C / BN);
        k_gemm_wmma_f32<<<grid, T, 0, stream>>>(bufB, W2, bufA, N, HID, C);
        k_selfloop_init<<<cdiv((long long)N * C / 4, T), T, 0, stream>>>(bufB, bufA, dinv, N, C);
        k_edge_scatter<<<cdiv(E, 8), T, 0, stream>>>(bufB, bufA, rowp, colp, dinv, E, C);
        k_bias_act<<<cdiv((long long)N * C / 4, T), T, 0, stream>>>(bufB, b2, N, C, 0);
        k_softmax64<<<cdiv(N, 8), T, 0, stream>>>(bufB, N);
    }

    // --- Layer 3: h3 = soft @ W3 ; agg -> d_out ; +b3 ---
    {
        dim3 grid(cdiv(N, BM), C / BN);
        k_gemm_wmma_f32<<<grid, T, 0, stream>>>(bufB, W3, bufA, N, C, C);
        k_selfloop_init<<<cdiv((long long)N * C / 4, T), T, 0, stream>>>(outp, bufA, dinv, N, C);
        k_edge_scatter<<<cdiv(E, 8), T, 0, stream>>>(outp, bufA, rowp, colp, dinv, E, C);
        k_bias_act<<<cdiv((long long)N * C / 4, T), T, 0, stream>>>(outp, b3, N, C, 0);
    }
}